// CenterLoss_80367428042769
// MI455X (gfx1250) — compile-verified
//
#include <hip/hip_runtime.h>
#include <hip/hip_bf16.h>
#include <cstdint>

// Problem constants (match reference)
#define BB 64
#define DD 128
#define TT 4096
#define CC 100
#define NN (BB * TT)            // 262144 tokens
#define TPB 256
#define NBLK 256                // stage-1 blocks
#define TOK_PER_BLK (NN / NBLK) // 1024 tokens per block (4 per thread via float4)
#define SSTRIDE 129             // LDS row pad: odd stride -> distinct banks per label

// Workspace layout (bytes):
//   P : NBLK * CC * DD floats   (per-block class sums)      13,107,200
//   H : NBLK * CC ints          (per-block label histogram)    102,400
//   L : NBLK floats             (per-block loss partials)        1,024
// total ~12.6 MiB required in d_ws.

typedef __attribute__((ext_vector_type(2))) float v2f;
typedef __attribute__((ext_vector_type(8))) float v8f;

#if defined(__gfx1250__) && __has_builtin(__builtin_amdgcn_global_load_async_to_lds_b32)
#define HAVE_ASYNC_LDS 1
typedef __attribute__((address_space(1))) int as1_int;
typedef __attribute__((address_space(3))) int as3_int;
#endif

__global__ __launch_bounds__(TPB) void cl_stage1(
    const float* __restrict__ feature, const float* __restrict__ centers,
    const int* __restrict__ labels, float* __restrict__ P,
    int* __restrict__ H, float* __restrict__ L)
{
    __shared__ float s_centers[CC * SSTRIDE];
    __shared__ float s_sums[CC * SSTRIDE];
    __shared__ int   s_hist[CC];
    __shared__ float s_red[TPB / 32];

    const int tid = threadIdx.x;
    const int blk = blockIdx.x;
    const int b   = blk >> 2;                  // 4 chunks of 1024 tokens per batch row
    const int t0  = (blk & 3) * TOK_PER_BLK;
    const int t   = t0 + tid * 4;

    // Zero LDS accumulators
    for (int i = tid; i < CC * SSTRIDE; i += TPB) s_sums[i] = 0.0f;
    for (int i = tid; i < CC; i += TPB) s_hist[i] = 0;

    // Stage centers into LDS with stride-129 padded rows.
    // CDNA5 path: async global->LDS copies overlapped with the norm pass below.
    for (int idx = tid; idx < CC * DD; idx += TPB) {
        const int c = idx >> 7;
        const int d = idx & 127;
#if HAVE_ASYNC_LDS
        __builtin_amdgcn_global_load_async_to_lds_b32(
            (as1_int*)(centers + idx),
            (as3_int*)&s_centers[c * SSTRIDE + d],
            0, 0);
#else
        s_centers[c * SSTRIDE + d] = centers[idx];
#endif
    }

    // Phase A: accumulate sum of squares over D for this thread's 4 tokens.
    // feature[b][d][t] layout: stride T between d's; float4 over t is coalesced.
    const float* fbase = feature + ((size_t)b * DD) * TT + t;
    float ss0 = 0.f, ss1 = 0.f, ss2 = 0.f, ss3 = 0.f;
#pragma unroll 8
    for (int d = 0; d < DD; ++d) {
        const float4 v = *(const float4*)(fbase + (size_t)d * TT);
        ss0 += v.x * v.x; ss1 += v.y * v.y;
        ss2 += v.z * v.z; ss3 += v.w * v.w;
    }
    const float in0 = 1.0f / fmaxf(sqrtf(ss0), 1e-12f);
    const float in1 = 1.0f / fmaxf(sqrtf(ss1), 1e-12f);
    const float in2 = 1.0f / fmaxf(sqrtf(ss2), 1e-12f);
    const float in3 = 1.0f / fmaxf(sqrtf(ss3), 1e-12f);

    const int4 lb = *(const int4*)(labels + b * TT + t);
    const int o0 = lb.x * SSTRIDE, o1 = lb.y * SSTRIDE;
    const int o2 = lb.z * SSTRIDE, o3 = lb.w * SSTRIDE;

#if HAVE_ASYNC_LDS
#if __has_builtin(__builtin_amdgcn_s_wait_asynccnt)
    __builtin_amdgcn_s_wait_asynccnt(0);
#else
    asm volatile("s_wait_asynccnt 0" ::: "memory");
#endif
#endif
    __syncthreads();

    // Phase B: second pass (L2-hot), compute diff, loss and LDS scatter-add.
    float loss = 0.0f;
#pragma unroll 4
    for (int d = 0; d < DD; ++d) {
        const float4 v = *(const float4*)(fbase + (size_t)d * TT);
        float f0 = v.x * in0, c0 = s_centers[o0 + d], d0 = c0 - f0;
        float f1 = v.y * in1, c1 = s_centers[o1 + d], d1 = c1 - f1;
        float f2 = v.z * in2, c2 = s_centers[o2 + d], d2 = c2 - f2;
        float f3 = v.w * in3, c3 = s_centers[o3 + d], d3 = c3 - f3;
        loss += d0 * d0; loss += d1 * d1; loss += d2 * d2; loss += d3 * d3;
        atomicAdd(&s_sums[o0 + d], d0);
        atomicAdd(&s_sums[o1 + d], d1);
        atomicAdd(&s_sums[o2 + d], d2);
        atomicAdd(&s_sums[o3 + d], d3);
    }
    atomicAdd(&s_hist[lb.x], 1);
    atomicAdd(&s_hist[lb.y], 1);
    atomicAdd(&s_hist[lb.z], 1);
    atomicAdd(&s_hist[lb.w], 1);

    // Loss: wave32 reduce, then block reduce.
    const int lane = tid & 31, wv = tid >> 5;
    float l = loss;
    for (int m = 16; m; m >>= 1) l += __shfl_xor(l, m, 32);
    if (lane == 0) s_red[wv] = l;
    __syncthreads();   // also fences all LDS atomics before flush
    if (tid == 0) {
        float tot = 0.f;
        for (int i = 0; i < TPB / 32; ++i) tot += s_red[i];
        L[blk] = tot;
    }

    // Flush per-block partials to workspace (de-pad stride 129 -> 128).
    float* Pb = P + (size_t)blk * (CC * DD);
    for (int i = tid; i < CC * DD; i += TPB)
        Pb[i] = s_sums[(i >> 7) * SSTRIDE + (i & 127)];
    int* Hb = H + blk * CC;
    for (int i = tid; i < CC; i += TPB) Hb[i] = s_hist[i];
}

// Stage 2: deterministic cross-block reduction.
// difference[c,d] = (sum over 256 block partials) / max(count[c],1)
// The 256-way sum is done with V_WMMA_F32_16X16X4_F32 using an all-ones A
// matrix: D[m,n] = sum_k B[k,n] + C[m,n], i.e. each WMMA folds 4 partial rows
// in exact f32 with a fixed accumulation order. One wave owns 16 outputs.
__global__ __launch_bounds__(256) void cl_stage2(
    const float* __restrict__ P, const int* __restrict__ H,
    const float* __restrict__ L, float* __restrict__ out)
{
    if (blockIdx.x == CC) {  // loss-reduction block
        __shared__ float sr[256];
        sr[threadIdx.x] = L[threadIdx.x];  // NBLK == 256
        __syncthreads();
        for (int s = 128; s > 0; s >>= 1) {
            if (threadIdx.x < s) sr[threadIdx.x] += sr[threadIdx.x + s];
            __syncthreads();
        }
        if (threadIdx.x == 0)
            out[0] = sr[0] * (1.0f / (float)((long long)NN * DD));
        return;
    }

    const int wv   = threadIdx.x >> 5;
    const int lane = threadIdx.x & 31;
    const int tile = blockIdx.x * 8 + wv;        // 0..799 (800 tiles of 16 outputs)
    const int col  = tile * 16 + (lane & 15);    // flattened (c*128 + d) index
    const int hv   = lane >> 4;

    v2f a; a.x = 1.0f; a.y = 1.0f;               // all-ones 16x4 A fragment
    v8f acc = {};
    for (int g = 0; g < NBLK / 4; ++g) {
        v2f bv;
        bv.x = P[(size_t)(g * 4 + hv)     * (CC * DD) + col];
        bv.y = P[(size_t)(g * 4 + 2 + hv) * (CC * DD) + col];
        // D = ones(16x4) * B(4x16) + C : sums 4 block-partials per issue.
        acc = __builtin_amdgcn_wmma_f32_16x16x4_f32(
            false, a, false, bv, (short)0, acc, false, false);
    }

    // All 16 columns of this tile belong to one class (128 % 16 == 0).
    const int cls = (tile * 16) / DD;
    float cnt = 0.0f;
    for (int blk = 0; blk < NBLK; ++blk) cnt += (float)H[blk * CC + cls];
    cnt = fmaxf(cnt, 1.0f);

    if (lane < 16)                    // row M=0 lives in VGPR0 of lanes 0..15
        out[1 + col] = acc[0] / cnt;
}

extern "C" void kernel_launch(void* const* d_in, const int* in_sizes, int n_in,
                              void* d_out, int out_size, void* d_ws, size_t ws_size,
                              hipStream_t stream)
{
    (void)in_sizes; (void)n_in; (void)out_size; (void)ws_size;

    const float* feature = (const float*)d_in[0];
    const float* cls     = (const float*)d_in[1];
    const float* centers = (const float*)d_in[2];
    const int*   labels  = (const int*)d_in[3];
    float* out = (float*)d_out;

    float* P = (float*)d_ws;
    int*   H = (int*)((char*)d_ws + (size_t)NBLK * CC * DD * sizeof(float));
    float* L = (float*)((char*)d_ws + (size_t)NBLK * CC * DD * sizeof(float)
                                    + (size_t)NBLK * CC * sizeof(int));

    cl_stage1<<<NBLK, TPB, 0, stream>>>(feature, centers, labels, P, H, L);
    cl_stage2<<<CC + 1, 256, 0, stream>>>(P, H, L, out);

    // Pass-through outputs: cls (6400 f32) and labels (262144 i32, bit-copied).
    (void)hipMemcpyAsync(out + 1 + CC * DD, cls, (size_t)BB * CC * sizeof(float),
                         hipMemcpyDeviceToDevice, stream);
    (void)hipMemcpyAsync(out + 1 + CC * DD + BB * CC, labels,
                         (size_t)BB * TT * sizeof(int),
                         hipMemcpyDeviceToDevice, stream);
}